// AdditiveAttentionScoreOp_27608049779401
// MI455X (gfx1250) — compile-verified
//
#include <hip/hip_runtime.h>
#include <hip/hip_bf16.h>

#define BH 8      // B*H heads
#define LL 512    // sequence length
#define DD 64     // feature dim
#define HID 64    // hidden dim

typedef __attribute__((ext_vector_type(2))) float v2f;
typedef __attribute__((ext_vector_type(8))) float v8f;

__device__ __forceinline__ float fast_tanh(float x) {
#if __has_builtin(__builtin_amdgcn_tanhf)
    return __builtin_amdgcn_tanhf(x);
#else
    float r;
    asm volatile("v_tanh_f32 %0, %1" : "=v"(r) : "v"(x));
    return r;
#endif
}

// ---------------------------------------------------------------------------
// Projection kernel: q_proj = Q * W1^T (row-major [H][L][HID])
//                    k_projT = (K * W2^T)^T (stored [H][HID][L] for coalesced
//                    scoring-loop loads)
// One wave32 computes one 16x16 fp32 tile via V_WMMA_F32_16X16X4_F32,
// accumulating 16 K-steps (K=64).
// Tile index space: proj(2) x head(8) x mtile(32) x ntile(4) = 2048 waves.
// ---------------------------------------------------------------------------
__global__ void proj_wmma_kernel(const float* __restrict__ q,
                                 const float* __restrict__ k,
                                 const float* __restrict__ W1,
                                 const float* __restrict__ W2,
                                 float* __restrict__ qp,    // [BH][LL][HID]
                                 float* __restrict__ kpt)   // [BH][HID][LL]
{
    const int wave = (blockIdx.x * blockDim.x + threadIdx.x) >> 5;
    const int lane = threadIdx.x & 31;

    const int ntile = wave & 3;          // 4 tiles over HID=64
    const int mtile = (wave >> 2) & 31;  // 32 tiles over L=512
    const int head  = (wave >> 7) & 7;
    const int proj  = wave >> 10;        // 0 -> q/W1, 1 -> k/W2

    const float* __restrict__ A = (proj == 0 ? q : k) + head * LL * DD; // [512][64]
    const float* __restrict__ W = (proj == 0 ? W1 : W2);                // [HID][DD]

    const int m0   = mtile * 16;
    const int n0   = ntile * 16;
    const int half = lane >> 4;   // 0: K pair {k,k+1}, 1: {k+2,k+3}
    const int l15  = lane & 15;
    const int arow = m0 + l15;    // A row (M index) for this lane
    const int wrow = n0 + l15;    // W row = output channel (N index)

    v8f c = {};
#pragma unroll
    for (int kk = 0; kk < DD; kk += 4) {
        const int ke = kk + half * 2;
        v2f a, b;
        a.x = A[arow * DD + ke];
        a.y = A[arow * DD + ke + 1];
        // B[d][e] = W[e][d]  (B = W^T), column e lives in lane (e & 15)
        b.x = W[wrow * DD + ke];
        b.y = W[wrow * DD + ke + 1];
        c = __builtin_amdgcn_wmma_f32_16x16x4_f32(
                /*neg_a=*/false, a, /*neg_b=*/false, b,
                /*c_mod=*/(short)0, c, /*reuse_a=*/false, /*reuse_b=*/false);
    }

    // C/D layout: VGPR r -> M = m0 + half*8 + r, N = wrow
    if (proj == 0) {
        float* __restrict__ o = qp + head * LL * HID;
#pragma unroll
        for (int r = 0; r < 8; ++r)
            o[(m0 + half * 8 + r) * HID + wrow] = c[r];
    } else {
        float* __restrict__ o = kpt + head * HID * LL;
#pragma unroll
        for (int r = 0; r < 8; ++r)
            o[wrow * LL + (m0 + half * 8 + r)] = c[r];
    }
}

// ---------------------------------------------------------------------------
// Scoring kernel: out[h][q][k] = sum_e v[e] * tanh(qp[h][q][e] + kpt[h][e][k])
// One 256-thread block per (h, q); each thread produces 2 adjacent k outputs
// using float2 loads (fully coalesced: kpt is e-major). tanh uses the CDNA5
// hardware transcendental.
// ---------------------------------------------------------------------------
__global__ void score_kernel(const float* __restrict__ qp,   // [BH][LL][HID]
                             const float* __restrict__ kpt,  // [BH][HID][LL]
                             const float* __restrict__ v,    // [HID]
                             float* __restrict__ out)        // [BH][LL][LL]
{
    const int h  = blockIdx.x >> 9;     // / 512
    const int qi = blockIdx.x & (LL - 1);
    const int tid = threadIdx.x;        // 0..255

    __shared__ float qv[HID];
    __shared__ float vv[HID];
    if (tid < HID) {
        qv[tid] = qp[(h * LL + qi) * HID + tid];
        vv[tid] = v[tid];
    }
    __syncthreads();

    const float* __restrict__ kp = kpt + h * HID * LL;
    const int k0 = tid * 2;

    float acc0 = 0.0f, acc1 = 0.0f;
#pragma unroll 8
    for (int e = 0; e < HID; ++e) {
        const float2 kk = *(const float2*)(kp + e * LL + k0);
        const float qe = qv[e];
        const float ve = vv[e];
        acc0 = __builtin_fmaf(ve, fast_tanh(qe + kk.x), acc0);
        acc1 = __builtin_fmaf(ve, fast_tanh(qe + kk.y), acc1);
    }

    float2* o = (float2*)(out + (h * LL + qi) * LL + k0);
    *o = make_float2(acc0, acc1);
}

extern "C" void kernel_launch(void* const* d_in, const int* in_sizes, int n_in,
                              void* d_out, int out_size, void* d_ws, size_t ws_size,
                              hipStream_t stream) {
    const float* q  = (const float*)d_in[0];   // [1,8,512,64]
    const float* k  = (const float*)d_in[1];   // [1,8,512,64]
    const float* W1 = (const float*)d_in[2];   // [64,64]
    const float* W2 = (const float*)d_in[3];   // [64,64]
    const float* v  = (const float*)d_in[4];   // [1,64]
    float* out = (float*)d_out;                // [1,8,512,512]

    float* qp  = (float*)d_ws;                 // 8*512*64 floats = 1 MB
    float* kpt = qp + BH * LL * HID;           // 8*64*512 floats = 1 MB

    // 2048 wave-tiles, 8 waves (256 threads) per block -> 256 blocks
    proj_wmma_kernel<<<256, 256, 0, stream>>>(q, k, W1, W2, qp, kpt);

    // one block per (head, q-row): 8*512 = 4096 blocks
    score_kernel<<<BH * LL, 256, 0, stream>>>(qp, kpt, v, out);
}